// ConditionalUvTextureImagePlus2dHistoCritic_56727928045829
// MI455X (gfx1250) — compile-verified
//
#include <hip/hip_runtime.h>
#include <hip/hip_bf16.h>
#include <math.h>

typedef __attribute__((ext_vector_type(2))) float v2f;
typedef __attribute__((ext_vector_type(8))) float v8f;

// ---------------------------------------------------------------------------
// Kernel 1: per-batch sigma = clamp(mean_over_channels(std(T, ddof=1)), bd)
// One block (256 threads) per batch.
// ---------------------------------------------------------------------------
__global__ void kde_sigma_kernel(const float* __restrict__ T,
                                 float* __restrict__ sigma_ws,
                                 int nV, float bd) {
    const int b   = blockIdx.x;
    const int tid = threadIdx.x;
    const float* Tb = T + (size_t)b * nV * 2;

    float sx = 0.f, sy = 0.f, ssx = 0.f, ssy = 0.f;
    for (int i = tid; i < nV; i += 256) {
        float x = Tb[2 * i + 0];
        float y = Tb[2 * i + 1];
        sx += x; sy += y; ssx += x * x; ssy += y * y;
    }

    __shared__ float rx[256], ry[256], rxx[256], ryy[256];
    rx[tid] = sx; ry[tid] = sy; rxx[tid] = ssx; ryy[tid] = ssy;
    __syncthreads();
    for (int ofs = 128; ofs > 0; ofs >>= 1) {
        if (tid < ofs) {
            rx[tid]  += rx[tid + ofs];
            ry[tid]  += ry[tid + ofs];
            rxx[tid] += rxx[tid + ofs];
            ryy[tid] += ryy[tid + ofs];
        }
        __syncthreads();
    }
    if (tid == 0) {
        float n  = (float)nV;
        float vx = (rxx[0] - rx[0] * rx[0] / n) / (n - 1.0f);
        float vy = (ryy[0] - ry[0] * ry[0] / n) / (n - 1.0f);
        float sig = 0.5f * (sqrtf(fmaxf(vx, 0.0f)) + sqrtf(fmaxf(vy, 0.0f)));
        sigma_ws[b] = fmaxf(sig, bd);
    }
}

// ---------------------------------------------------------------------------
// Kernel 2: main KDE accumulation via V_WMMA_F32_16X16X4_F32.
//   D = A x B with A row m = [Sx, Sy, s2, 1], B col n = [-2Tx, -2Ty, 1, t2]
//   => D[m][n] = ||S[m] - T[n]||^2 exactly, one WMMA per 16x16 tile.
// Block = 256 threads = 8 waves; block handles one (batch, 16-sample tile);
// wave w strides over 16-vertex chunks w, w+8, w+16, ...
// The chunk index is forced into SGPRs (readfirstlane) and the software
// pipeline is branchless (clamped scalar selects), so the hot loop has no
// EXEC manipulation and keeps one global_load_b64 in flight.
// Ragged tail (nV % 16) is handled by wave 0 only.
// ---------------------------------------------------------------------------
__global__ void __launch_bounds__(256)
kde_wmma_kernel(const float* __restrict__ T, const float* __restrict__ S,
                const float* __restrict__ sigma_ws, float* __restrict__ kh,
                int nV, int nS, float cf, float cn) {
    const int nTiles = nS >> 4;            // sample tiles of 16
    const int b      = blockIdx.x / nTiles;
    const int st     = blockIdx.x % nTiles;
    const int tid    = threadIdx.x;
    // wave index is uniform across the wave: pin it to an SGPR so the chunk
    // loop and pipeline guards compile to scalar control flow.
    const int wv     = __builtin_amdgcn_readfirstlane(tid >> 5);
    const int lane   = tid & 31;
    const int half   = lane >> 4;          // 0: K=0/1 rows, 1: K=2/3 rows
    const int idx16  = lane & 15;

    const float sigma  = sigma_ws[b];
    // fold exp -> exp2: E = exp2( (cn/sigma^2 * log2(e)) * D )
    const float escale = (cn / (sigma * sigma)) * 1.44269504088896340736f; // < 0

    // --- A matrix (loop-invariant): lane = M = idx16 -------------------------
    const int   sidx = st * 16 + idx16;
    const float sx   = S[2 * sidx + 0];
    const float sy   = S[2 * sidx + 1];
    const float s2   = sx * sx + sy * sy;
    v2f a;
    a.x = half ? s2   : sx;   // VGPR0: K=2 | K=0
    a.y = half ? 1.0f : sy;   // VGPR1: K=3 | K=1

    const float* Tb = T + (size_t)b * nV * 2;
    // per-lane base; chunk ch starts at float offset ch*32
    const float* laneBase = Tb + 2 * idx16;

    v8f acc = {0.f, 0.f, 0.f, 0.f, 0.f, 0.f, 0.f, 0.f};
    const v8f czero = {0.f, 0.f, 0.f, 0.f, 0.f, 0.f, 0.f, 0.f};

    // ---- main loop: full 16-vertex chunks, branchless pipeline --------------
    const int fullChunks = nV >> 4;        // chunks entirely in range
    float tx = 0.f, ty = 0.f;
    if (wv < fullChunks) {                 // scalar branch (prologue load)
        tx = laneBase[wv * 32 + 0];
        ty = laneBase[wv * 32 + 1];
    }
    for (int ch = wv; ch < fullChunks; ch += 8) {
        // next chunk, clamped in-range (scalar select, no branch)
        const int nch = ch + 8;
        const int lch = (nch < fullChunks) ? nch : ch;
        const float ntx = laneBase[lch * 32 + 0];
        const float nty = laneBase[lch * 32 + 1];
        // warm the chunk two strides out (clamped; prefetch is side-effect free)
        const int pch = (ch + 16 < fullChunks) ? ch + 16 : lch;
        __builtin_prefetch(&laneBase[pch * 32], 0, 1);

        // --- process current chunk (EXEC all-1s throughout) -----------------
        const float t2 = tx * tx + ty * ty;
        v2f bm;
        bm.x = half ? 1.0f : -2.0f * tx;   // VGPR0: K=2 | K=0
        bm.y = half ? t2   : -2.0f * ty;   // VGPR1: K=3 | K=1

        v8f dmat = __builtin_amdgcn_wmma_f32_16x16x4_f32(
            /*neg_a=*/false, a, /*neg_b=*/false, bm,
            /*c_mod=*/(short)0, czero, /*reuse_a=*/false, /*reuse_b=*/false);

#pragma unroll
        for (int r = 0; r < 8; ++r)
            acc[r] += __builtin_amdgcn_exp2f(escale * dmat[r]); // v_exp_f32

        tx = ntx; ty = nty;
    }

    // ---- ragged tail (nV % 16 != 0): wave 0 only, predicated loads ----------
    if ((nV & 15) && wv == 0) {            // scalar entry, EXEC full
        const int v = fullChunks * 16 + idx16;
        float ttx = 0.f, tty = 0.f, tt2 = 1e30f;  // OOB -> exp2(-huge) == 0
        if (v < nV) {
            ttx = Tb[2 * v + 0];
            tty = Tb[2 * v + 1];
            tt2 = ttx * ttx + tty * tty;
        }
        v2f bm;
        bm.x = half ? 1.0f : -2.0f * ttx;
        bm.y = half ? tt2  : -2.0f * tty;
        v8f dmat = __builtin_amdgcn_wmma_f32_16x16x4_f32(
            false, a, false, bm, (short)0, czero, false, false);
#pragma unroll
        for (int r = 0; r < 8; ++r)
            acc[r] += __builtin_amdgcn_exp2f(escale * dmat[r]);
    }

    // --- reduce over columns (N) and waves ----------------------------------
    __shared__ float red[8][16][16];   // [wave][sample][column] = 8 KB
#pragma unroll
    for (int r = 0; r < 8; ++r)
        red[wv][r + 8 * half][idx16] = acc[r];
    __syncthreads();

    if (tid < 16) {
        float sum = 0.f;
        for (int w = 0; w < 8; ++w)
#pragma unroll
            for (int n = 0; n < 16; ++n)
                sum += red[w][tid][n];
        // K_H = cf * sigma^-d * mean_v(E), d = 2
        const float val = cf / (sigma * sigma) * (sum / (float)nV);
        kh[(size_t)b * nS + st * 16 + tid] = val;
    }
}

// ---------------------------------------------------------------------------
// Kernel 3: normalize K_H over the sample axis with clamp(sum, 1e-5).
// One block per batch.
// ---------------------------------------------------------------------------
__global__ void kde_norm_kernel(const float* __restrict__ kh,
                                float* __restrict__ out, int nS) {
    const int b   = blockIdx.x;
    const int tid = threadIdx.x;
    __shared__ float sdata[256];

    float s = 0.f;
    for (int i = tid; i < nS; i += 256) s += kh[(size_t)b * nS + i];
    sdata[tid] = s;
    __syncthreads();
    for (int ofs = 128; ofs > 0; ofs >>= 1) {
        if (tid < ofs) sdata[tid] += sdata[tid + ofs];
        __syncthreads();
    }
    const float denom = fmaxf(sdata[0], 1e-5f);
    for (int i = tid; i < nS; i += 256)
        out[(size_t)b * nS + i] = kh[(size_t)b * nS + i] / denom;
}

// ---------------------------------------------------------------------------
extern "C" void kernel_launch(void* const* d_in, const int* in_sizes, int n_in,
                              void* d_out, int out_size, void* d_ws, size_t ws_size,
                              hipStream_t stream) {
    const float* T = (const float*)d_in[0];   // [B, nV, 2] f32
    const float* S = (const float*)d_in[1];   // [nS, 2]    f32
    float* out = (float*)d_out;               // [B, nS]    f32

    const int nS = in_sizes[1] / 2;           // 1024
    const int B  = out_size / nS;             // 4
    const int nV = in_sizes[0] / (2 * B);     // 50000

    const int npd = (int)(sqrt((double)nS) + 0.5);   // 32
    const float bd = 1.0f / (float)(npd - 1);        // min sigma

    const double d  = 2.0;
    const double c  = pow(4.0 / (d + 2.0), 1.0 / (d + 4.0)) *
                      pow((double)nV, -1.0 / (d + 4.0));
    const float  cf = (float)(pow(2.0 * M_PI, -d / 2.0) / pow(c, d));
    const float  cn = (float)(-1.0 / (2.0 * c * c));

    float* sigma_ws = (float*)d_ws;           // [B]
    float* kh       = sigma_ws + 16;          // [B, nS] unnormalized K_H

    kde_sigma_kernel<<<B, 256, 0, stream>>>(T, sigma_ws, nV, bd);

    const int nTiles = nS / 16;               // 64
    kde_wmma_kernel<<<B * nTiles, 256, 0, stream>>>(T, S, sigma_ws, kh,
                                                    nV, nS, cf, cn);

    kde_norm_kernel<<<B, 256, 0, stream>>>(kh, out, nS);
}